// GATEncoder_22351009808429
// MI455X (gfx1250) — compile-verified
//
#include <hip/hip_runtime.h>
#include <math.h>

typedef __attribute__((ext_vector_type(2))) float v2f;
typedef __attribute__((ext_vector_type(8))) float v8f;

#define NNODE 50000
#define FIN   256
#define F1    256   // layer1 out features = 4 heads * 64
#define F2    64    // layer2 out features = 1 head * 64

#define TILE_M 128
#define TILE_N 64
#define TILE_K 32
#define A_STRIDE 36   // 32 + 4 pad: 16-row fragment reads hit distinct banks
#define B_STRIDE 80   // 64 + 16 pad: half-waves (k, k+2 rows) use disjoint bank ranges

__device__ __forceinline__ float lrelu(float v) { return v > 0.f ? v : 0.2f * v; }

// float atomic max with -inf init: positive -> signed int max, negative -> unsigned min
__device__ __forceinline__ void atomicMaxF(float* addr, float v) {
  if (v >= 0.f) atomicMax((int*)addr, __float_as_int(v));
  else          atomicMin((unsigned int*)addr, __float_as_uint(v));
}

// ---------------------------------------------------------------------------
// fill a float buffer with a constant (grid-stride)
__global__ void fill_kernel(float* p, long long n, float val) {
  long long i = (long long)blockIdx.x * blockDim.x + threadIdx.x;
  long long stride = (long long)gridDim.x * blockDim.x;
  for (; i < n; i += stride) p[i] = val;
}

// ---------------------------------------------------------------------------
// C[M x Nn] = A[M x K] * B[K x Nn], row-major fp32, via V_WMMA_F32_16X16X4_F32.
// Block = 256 threads = 8 waves; block tile 128(rows) x 64(cols); each wave
// computes a 16 x 64 strip. K staged through LDS in chunks of 32 with
// coalesced float4 cooperative loads. Fragment layouts per ISA 7.12.2:
//   A 16x4 f32: VGPR0 = K{0,2}, VGPR1 = K{1,3} across half-waves.
//   C/D 16x16:  8 VGPRs; lanes 0-15 rows v, lanes 16-31 rows v+8.
__global__ __launch_bounds__(256) void gemm_f32_wmma(
    const float* __restrict__ A, const float* __restrict__ B, float* __restrict__ C,
    int M, int K, int Nn) {
  __shared__ float As[TILE_M * A_STRIDE];   // 18 KB
  __shared__ float Bs[TILE_K * B_STRIDE];   // 10 KB

  const int tid    = threadIdx.x;
  const int wave   = tid >> 5;
  const int lane   = tid & 31;
  const int lmod   = lane & 15;
  const int klane  = lane >> 4;            // 0 or 1
  const int mBlock = blockIdx.x * TILE_M;
  const int n0     = blockIdx.y * TILE_N;
  const int m0     = mBlock + wave * 16;

  // cooperative-load mapping: A tile 128 rows x 32 floats (16 floats/thread)
  const int arow  = tid >> 1;
  const int ahalf = (tid & 1) * 16;
  // B tile 32 rows x 64 floats (threads 0..127, 16 floats/thread)
  const int brow  = tid >> 2;
  const int bseg  = (tid & 3) * 16;

  v8f acc0 = {}, acc1 = {}, acc2 = {}, acc3 = {};

  for (int k0 = 0; k0 < K; k0 += TILE_K) {
    // ---- stage A chunk ----
    {
      int rg = mBlock + arow;
      float4 z = make_float4(0.f, 0.f, 0.f, 0.f);
      float4 v0 = z, v1 = z, v2 = z, v3 = z;
      if (rg < M) {
        const float4* src = (const float4*)(A + (size_t)rg * K + k0 + ahalf);
        v0 = src[0]; v1 = src[1]; v2 = src[2]; v3 = src[3];
      }
      float4* dst = (float4*)(&As[arow * A_STRIDE + ahalf]);
      dst[0] = v0; dst[1] = v1; dst[2] = v2; dst[3] = v3;
    }
    // ---- stage B chunk ----
    if (tid < 128) {
      const float4* src = (const float4*)(B + (size_t)(k0 + brow) * Nn + n0 + bseg);
      float4 v0 = src[0], v1 = src[1], v2 = src[2], v3 = src[3];
      float4* dst = (float4*)(&Bs[brow * B_STRIDE + bseg]);
      dst[0] = v0; dst[1] = v1; dst[2] = v2; dst[3] = v3;
    }
    __syncthreads();

#pragma unroll
    for (int kk = 0; kk < TILE_K; kk += 4) {
      v2f a = *(const v2f*)(&As[(wave * 16 + lmod) * A_STRIDE + kk + 2 * klane]);
      int r0 = (kk + 2 * klane) * B_STRIDE;
      int r1 = r0 + B_STRIDE;
      v2f b0; b0.x = Bs[r0 +  0 + lmod]; b0.y = Bs[r1 +  0 + lmod];
      v2f b1; b1.x = Bs[r0 + 16 + lmod]; b1.y = Bs[r1 + 16 + lmod];
      v2f b2; b2.x = Bs[r0 + 32 + lmod]; b2.y = Bs[r1 + 32 + lmod];
      v2f b3; b3.x = Bs[r0 + 48 + lmod]; b3.y = Bs[r1 + 48 + lmod];
      acc0 = __builtin_amdgcn_wmma_f32_16x16x4_f32(false, a, false, b0, (short)0, acc0, false, false);
      acc1 = __builtin_amdgcn_wmma_f32_16x16x4_f32(false, a, false, b1, (short)0, acc1, false, false);
      acc2 = __builtin_amdgcn_wmma_f32_16x16x4_f32(false, a, false, b2, (short)0, acc2, false, false);
      acc3 = __builtin_amdgcn_wmma_f32_16x16x4_f32(false, a, false, b3, (short)0, acc3, false, false);
    }
    __syncthreads();
  }

  v8f* accs[4] = { &acc0, &acc1, &acc2, &acc3 };
#pragma unroll
  for (int t = 0; t < 4; ++t) {
#pragma unroll
    for (int v = 0; v < 8; ++v) {
      int r = m0 + v + 8 * klane;
      if (r < M) C[(size_t)r * Nn + n0 + t * 16 + lmod] = (*accs[t])[v];
    }
  }
}

// ---------------------------------------------------------------------------
// alpha_src[n,h] = sum_c h[n, h*ch + c] * att_s[h, c]; same for dst
__global__ void attn_project(const float* __restrict__ h,
                             const float* __restrict__ att_s,
                             const float* __restrict__ att_d,
                             float* __restrict__ as, float* __restrict__ ad,
                             int n_nodes, int heads, int ch) {
  int i = blockIdx.x * blockDim.x + threadIdx.x;
  if (i >= n_nodes * heads) return;
  int hh = i % heads;
  int n  = i / heads;
  const float* hp = h + (size_t)n * heads * ch + (size_t)hh * ch;
  const float* sa = att_s + (size_t)hh * ch;
  const float* da = att_d + (size_t)hh * ch;
  float s0 = 0.f, s1 = 0.f;
  for (int c = 0; c < ch; ++c) { float v = hp[c]; s0 += v * sa[c]; s1 += v * da[c]; }
  as[(size_t)n * heads + hh] = s0;
  ad[(size_t)n * heads + hh] = s1;
}

// ---------------------------------------------------------------------------
// pass 1: scatter-max of leaky_relu(as[src]+ad[dst]) into m[dst]
__global__ void edge_softmax_max(const long long* __restrict__ ei, long long E, long long Etot,
                                 const float* __restrict__ as, const float* __restrict__ ad,
                                 float* __restrict__ mbuf, int heads) {
  long long e = (long long)blockIdx.x * blockDim.x + threadIdx.x;
  if (e >= Etot) return;
  long long s = (e < E) ? ei[e]     : (e - E);
  long long d = (e < E) ? ei[E + e] : (e - E);
  for (int hh = 0; hh < heads; ++hh) {
    float v = lrelu(as[s * heads + hh] + ad[d * heads + hh]);
    atomicMaxF(&mbuf[d * heads + hh], v);
  }
}

// pass 2: scatter-sum of exp(e - m[dst]) into den[dst]
__global__ void edge_softmax_sum(const long long* __restrict__ ei, long long E, long long Etot,
                                 const float* __restrict__ as, const float* __restrict__ ad,
                                 const float* __restrict__ mbuf, float* __restrict__ den,
                                 int heads) {
  long long e = (long long)blockIdx.x * blockDim.x + threadIdx.x;
  if (e >= Etot) return;
  long long s = (e < E) ? ei[e]     : (e - E);
  long long d = (e < E) ? ei[E + e] : (e - E);
  for (int hh = 0; hh < heads; ++hh) {
    float v = lrelu(as[s * heads + hh] + ad[d * heads + hh]);
    atomicAdd(&den[d * heads + hh], expf(v - mbuf[d * heads + hh]));
  }
}

// ---------------------------------------------------------------------------
// layer-1 aggregation: one wave per edge, 4 heads x 64 ch = 256 f32.
// lane handles 8 consecutive channels (all within one head: head = lane>>3).
__global__ __launch_bounds__(256) void edge_aggregate1(
    const long long* __restrict__ ei, long long E, long long Etot,
    const float* __restrict__ as, const float* __restrict__ ad,
    const float* __restrict__ mbuf, const float* __restrict__ den,
    const float* __restrict__ hsrc, float* __restrict__ out) {
  long long gw = ((long long)blockIdx.x * blockDim.x + threadIdx.x) >> 5;
  int lane = threadIdx.x & 31;
  if (gw >= Etot) return;
  long long s = (gw < E) ? ei[gw]     : (gw - E);
  long long d = (gw < E) ? ei[E + gw] : (gw - E);
  int hh = lane >> 3;  // (lane*8)/64
  float v  = lrelu(as[s * 4 + hh] + ad[d * 4 + hh]);
  float ex = expf(v - mbuf[d * 4 + hh]);
  float alpha = ex / (den[d * 4 + hh] + 1e-16f);

  const float4* hp = (const float4*)(hsrc + (size_t)s * 256 + lane * 8);
  float4 p0 = hp[0], p1 = hp[1];
  float* op = out + (size_t)d * 256 + lane * 8;
  atomicAdd(op + 0, p0.x * alpha);
  atomicAdd(op + 1, p0.y * alpha);
  atomicAdd(op + 2, p0.z * alpha);
  atomicAdd(op + 3, p0.w * alpha);
  atomicAdd(op + 4, p1.x * alpha);
  atomicAdd(op + 5, p1.y * alpha);
  atomicAdd(op + 6, p1.z * alpha);
  atomicAdd(op + 7, p1.w * alpha);
}

// layer-2 aggregation: one wave per edge, 1 head x 64 ch; lane handles 2 ch.
__global__ __launch_bounds__(256) void edge_aggregate2(
    const long long* __restrict__ ei, long long E, long long Etot,
    const float* __restrict__ as, const float* __restrict__ ad,
    const float* __restrict__ mbuf, const float* __restrict__ den,
    const float* __restrict__ hsrc, float* __restrict__ out) {
  long long gw = ((long long)blockIdx.x * blockDim.x + threadIdx.x) >> 5;
  int lane = threadIdx.x & 31;
  if (gw >= Etot) return;
  long long s = (gw < E) ? ei[gw]     : (gw - E);
  long long d = (gw < E) ? ei[E + gw] : (gw - E);
  float v  = lrelu(as[s] + ad[d]);
  float ex = expf(v - mbuf[d]);
  float alpha = ex / (den[d] + 1e-16f);

  const float2* hp = (const float2*)(hsrc + (size_t)s * 64 + lane * 2);
  float2 p = *hp;
  float* op = out + (size_t)d * 64 + lane * 2;
  atomicAdd(op + 0, p.x * alpha);
  atomicAdd(op + 1, p.y * alpha);
}

// ---------------------------------------------------------------------------
// act[i] = elu(agg[i] + bias[i % F])
__global__ void bias_elu(const float* __restrict__ agg, const float* __restrict__ bias,
                         float* __restrict__ act, long long n, int F) {
  long long i = (long long)blockIdx.x * blockDim.x + threadIdx.x;
  if (i >= n) return;
  float x = agg[i] + bias[i % F];
  act[i] = x > 0.f ? x : (expf(x) - 1.f);
}

// out[i] = agg[i] + bias[i % F]   (layer-2: H=1, mean over heads == identity)
__global__ void bias_final(const float* __restrict__ agg, const float* __restrict__ bias,
                           float* __restrict__ out, long long n, int F) {
  long long i = (long long)blockIdx.x * blockDim.x + threadIdx.x;
  if (i >= n) return;
  out[i] = agg[i] + bias[i % F];
}

// ---------------------------------------------------------------------------
extern "C" void kernel_launch(void* const* d_in, const int* in_sizes, int n_in,
                              void* d_out, int out_size, void* d_ws, size_t ws_size,
                              hipStream_t stream) {
  (void)n_in; (void)out_size; (void)ws_size;
  const float*     x   = (const float*)d_in[0];
  const long long* ei  = (const long long*)d_in[1];   // [2, E] int64
  const float*     W1  = (const float*)d_in[2];       // [256, 256]
  const float*     as1 = (const float*)d_in[3];       // [4, 64]
  const float*     ad1 = (const float*)d_in[4];
  const float*     b1  = (const float*)d_in[5];       // [256]
  const float*     W2  = (const float*)d_in[6];       // [256, 64]
  const float*     as2 = (const float*)d_in[7];       // [1, 64]
  const float*     ad2 = (const float*)d_in[8];
  const float*     b2  = (const float*)d_in[9];       // [64]

  const long long E    = (long long)in_sizes[1] / 2;
  const long long Etot = E + NNODE;                   // + self loops

  // workspace layout (floats)
  float* ws = (float*)d_ws;
  size_t o = 0;
  float* h1   = ws + o; o += (size_t)NNODE * F1;      // x@W1 ; later reused as elu activations
  float* out1 = ws + o; o += (size_t)NNODE * F1;      // layer-1 aggregation
  float* h2   = out1;                                 // reuse (out1 dead after bias_elu)
  float* out2 = out1 + (size_t)NNODE * F2;            // layer-2 aggregation
  float* a_s1 = ws + o; o += (size_t)NNODE * 4;
  float* a_d1 = ws + o; o += (size_t)NNODE * 4;
  float* m1   = ws + o; o += (size_t)NNODE * 4;
  float* den1 = ws + o; o += (size_t)NNODE * 4;
  float* a_s2 = ws + o; o += (size_t)NNODE;
  float* a_d2 = ws + o; o += (size_t)NNODE;
  float* m2   = ws + o; o += (size_t)NNODE;
  float* den2 = ws + o; o += (size_t)NNODE;

  const int TB = 256;
  const long long nF1 = (long long)NNODE * F1;
  const long long nF2 = (long long)NNODE * F2;

  // ---- init layer-1 accumulators (every call: graph replay determinism) ----
  fill_kernel<<<2048, TB, 0, stream>>>(out1, nF1, 0.f);
  fill_kernel<<<512,  TB, 0, stream>>>(m1,   (long long)NNODE * 4, -INFINITY);
  fill_kernel<<<512,  TB, 0, stream>>>(den1, (long long)NNODE * 4, 0.f);

  // ---- layer 1 ----
  dim3 g1((NNODE + TILE_M - 1) / TILE_M, F1 / TILE_N);
  gemm_f32_wmma<<<g1, 256, 0, stream>>>(x, W1, h1, NNODE, FIN, F1);

  attn_project<<<(NNODE * 4 + TB - 1) / TB, TB, 0, stream>>>(h1, as1, ad1, a_s1, a_d1, NNODE, 4, 64);

  long long egrid = (Etot + TB - 1) / TB;
  edge_softmax_max<<<(unsigned)egrid, TB, 0, stream>>>(ei, E, Etot, a_s1, a_d1, m1, 4);
  edge_softmax_sum<<<(unsigned)egrid, TB, 0, stream>>>(ei, E, Etot, a_s1, a_d1, m1, den1, 4);

  long long agrid = (Etot * 32 + TB - 1) / TB;  // one wave per edge
  edge_aggregate1<<<(unsigned)agrid, TB, 0, stream>>>(ei, E, Etot, a_s1, a_d1, m1, den1, h1, out1);

  bias_elu<<<(unsigned)((nF1 + TB - 1) / TB), TB, 0, stream>>>(out1, b1, h1, nF1, F1);

  // ---- init layer-2 accumulators ----
  fill_kernel<<<1024, TB, 0, stream>>>(out2, nF2, 0.f);
  fill_kernel<<<256,  TB, 0, stream>>>(m2,   (long long)NNODE, -INFINITY);
  fill_kernel<<<256,  TB, 0, stream>>>(den2, (long long)NNODE, 0.f);

  // ---- layer 2 ----
  dim3 g2((NNODE + TILE_M - 1) / TILE_M, F2 / TILE_N);
  gemm_f32_wmma<<<g2, 256, 0, stream>>>(h1, W2, h2, NNODE, F1, F2);

  attn_project<<<(NNODE + TB - 1) / TB, TB, 0, stream>>>(h2, as2, ad2, a_s2, a_d2, NNODE, 1, 64);

  edge_softmax_max<<<(unsigned)egrid, TB, 0, stream>>>(ei, E, Etot, a_s2, a_d2, m2, 1);
  edge_softmax_sum<<<(unsigned)egrid, TB, 0, stream>>>(ei, E, Etot, a_s2, a_d2, m2, den2, 1);
  edge_aggregate2<<<(unsigned)agrid, TB, 0, stream>>>(ei, E, Etot, a_s2, a_d2, m2, den2, h2, out2);

  bias_final<<<(unsigned)((nF2 + TB - 1) / TB), TB, 0, stream>>>(out2, b2, (float*)d_out, nF2, F2);
}